// GNNDis_57329223467536
// MI455X (gfx1250) — compile-verified
//
#include <hip/hip_runtime.h>
#include <math.h>

#define NN 768
#define FF 16
#define EMBD 32
#define HU 64
#define HE 68
#define GAMMA 0.9f

typedef __attribute__((ext_vector_type(2))) float v2f;
typedef __attribute__((ext_vector_type(8))) float v8f;

// ---- workspace layout (floats) ----
constexpr int OFF_HG   = 0;
constexpr int OFF_HL   = OFF_HG + NN * EMBD;
constexpr int OFF_SG   = OFF_HL + NN * EMBD;
constexpr int OFF_DG   = OFF_SG + NN;
constexpr int OFF_SL   = OFF_DG + NN;
constexpr int OFF_DL   = OFF_SL + NN;
constexpr int OFF_ATTG = OFF_DL + NN;
constexpr int OFF_ATTL = OFF_ATTG + NN * NN;
constexpr int OFF_YG   = OFF_ATTL + NN * NN;
constexpr int OFF_YL   = OFF_YG + NN * EMBD;
constexpr int OFF_UR   = OFF_YL + NN * EMBD;
constexpr int OFF_UC   = OFF_UR + NN * HU;
constexpr int OFF_ER   = OFF_UC + NN * HU;
constexpr int OFF_EC   = OFF_ER + NN * HE;
constexpr int OFF_VAL  = OFF_EC + NN * HE;

__device__ __forceinline__ float softplusf(float x) {
    return fmaxf(x, 0.0f) + log1pf(expf(-fabsf(x)));
}
__device__ __forceinline__ float reluf(float x) { return fmaxf(x, 0.0f); }

// ---------------- Kernel 1: x@W projection + attention scalars ----------------
__global__ void k_proj(const float* __restrict__ x,
                       const float* __restrict__ Wg, const float* __restrict__ asg,
                       const float* __restrict__ adg,
                       const float* __restrict__ Wl, const float* __restrict__ asl,
                       const float* __restrict__ adl,
                       float* __restrict__ ws) {
    int i = blockIdx.x * blockDim.x + threadIdx.x;
    if (i >= NN) return;
    float xi[FF];
#pragma unroll
    for (int f = 0; f < FF; ++f) xi[f] = x[i * FF + f];

    float sg = 0.f, dg = 0.f, sl = 0.f, dl = 0.f;
#pragma unroll 4
    for (int e = 0; e < EMBD; ++e) {
        float hg = 0.f, hl = 0.f;
#pragma unroll
        for (int f = 0; f < FF; ++f) {
            hg += xi[f] * Wg[f * EMBD + e];
            hl += xi[f] * Wl[f * EMBD + e];
        }
        ws[OFF_HG + i * EMBD + e] = hg;
        ws[OFF_HL + i * EMBD + e] = hl;
        sg += hg * asg[e]; dg += hg * adg[e];
        sl += hl * asl[e]; dl += hl * adl[e];
    }
    ws[OFF_SG + i] = sg; ws[OFF_DG + i] = dg;
    ws[OFF_SL + i] = sl; ws[OFF_DL + i] = dl;
}

// ---------------- Kernel 2: masked leaky-relu logits + row softmax ----------------
__global__ void k_att(const float* __restrict__ adj, float* __restrict__ ws) {
    const int i   = blockIdx.x;
    const int sel = blockIdx.y;            // 0 = global GAT, 1 = local GAT
    const int tid = threadIdx.x;           // 256 threads, 3 cols each
    __shared__ float sred[256];

    const float  s    = ws[(sel ? OFF_SL : OFF_SG) + i];
    const float* d    = ws + (sel ? OFF_DL : OFF_DG);
    float*       att  = ws + (sel ? OFF_ATTL : OFF_ATTG) + (size_t)i * NN;
    const float* arow = adj + (size_t)i * NN;

    float ev[3]; float msk[3];
    float lmax = -1e30f;
#pragma unroll
    for (int t = 0; t < 3; ++t) {
        int j = tid + t * 256;
        float a = arow[j];
        float e = s + d[j];
        e = (e > 0.f) ? e : 0.2f * e;      // leaky_relu(0.2)
        msk[t] = (a > 0.f) ? 1.f : 0.f;
        ev[t]  = (a > 0.f) ? e : -1e30f;
        lmax = fmaxf(lmax, ev[t]);
    }
    sred[tid] = lmax; __syncthreads();
    for (int st = 128; st > 0; st >>= 1) {
        if (tid < st) sred[tid] = fmaxf(sred[tid], sred[tid + st]);
        __syncthreads();
    }
    const float mx = sred[0]; __syncthreads();

    float ex[3]; float lsum = 0.f;
#pragma unroll
    for (int t = 0; t < 3; ++t) {
        ex[t] = (msk[t] > 0.f) ? expf(ev[t] - mx) : 0.f;
        lsum += ex[t];
    }
    sred[tid] = lsum; __syncthreads();
    for (int st = 128; st > 0; st >>= 1) {
        if (tid < st) sred[tid] += sred[tid + st];
        __syncthreads();
    }
    const float inv = 1.0f / sred[0];
#pragma unroll
    for (int t = 0; t < 3; ++t) att[tid + t * 256] = ex[t] * inv;
}

// ---------------- Kernel 3: y = att @ h via V_WMMA_F32_16X16X4_F32 ----------------
// One wave per 16x16 D tile. EXEC is all ones (32 uniform threads, no divergence).
__global__ void k_gemm_wmma(float* __restrict__ ws) {
    const int tm  = blockIdx.x;            // 48 row tiles
    const int tn  = blockIdx.y;            // 2 col tiles (EMB=32)
    const int sel = blockIdx.z;            // which GAT
    const float* att = ws + (sel ? OFF_ATTL : OFF_ATTG);
    const float* h   = ws + (sel ? OFF_HL  : OFF_HG);
    float*       y   = ws + (sel ? OFF_YL  : OFF_YG);

    const unsigned lane = threadIdx.x;     // 0..31
    const int half = lane >> 4;            // 0: lanes 0-15, 1: lanes 16-31
    const int l    = lane & 15;
    const int arow = tm * 16 + l;          // A: M = lane%16 in both halves
    const int bcol = tn * 16 + l;          // B/D: N = lane%16

    v8f c = {};
    for (int k0 = 0; k0 < NN; k0 += 4) {
        const int ka = k0 + half * 2;      // A v0/v1 hold K, K+1 (half-split at K=2)
        __builtin_prefetch(&att[(size_t)arow * NN + k0 + 32], 0, 1);
        v2f a, b;
        a.x = att[(size_t)arow * NN + ka];
        a.y = att[(size_t)arow * NN + ka + 1];
        b.x = h[(ka)     * EMBD + bcol];   // B 4x16: v0 lanes0-15=K0 / lanes16-31=K2
        b.y = h[(ka + 1) * EMBD + bcol];   //         v1 lanes0-15=K1 / lanes16-31=K3
        c = __builtin_amdgcn_wmma_f32_16x16x4_f32(
                /*neg_a=*/false, a, /*neg_b=*/false, b,
                /*c_mod=*/(short)0, c, /*reuse_a=*/false, /*reuse_b=*/false);
    }
#pragma unroll
    for (int r = 0; r < 8; ++r) {          // D: VGPR r -> M=r (lanes<16) / M=8+r
        int m = tm * 16 + r + (half ? 8 : 0);
        y[(size_t)m * EMBD + tn * 16 + l] = c[r];
    }
}

// ---------------- Kernel 4: per-node heads: val, ur/uc, er/ec ----------------
__global__ void k_heads(const float* __restrict__ uW1,
                        const float* __restrict__ eW1,
                        const float* __restrict__ vW1, const float* __restrict__ vb1,
                        const float* __restrict__ vW2, const float* __restrict__ vb2,
                        float* __restrict__ ws) {
    int i = blockIdx.x * blockDim.x + threadIdx.x;
    if (i >= NN) return;
    float yg[EMBD], yl[EMBD];
#pragma unroll
    for (int k = 0; k < EMBD; ++k) {
        yg[k] = ws[OFF_YG + i * EMBD + k];
        yl[k] = ws[OFF_YL + i * EMBD + k];
    }
    // value head: -softplus(relu(yg@vW1+vb1)@vW2+vb2)
    float acc = vb2[0];
    for (int h = 0; h < HU; ++h) {
        float t = vb1[h];
#pragma unroll
        for (int k = 0; k < EMBD; ++k) t += yg[k] * vW1[k * HU + h];
        acc += reluf(t) * vW2[h];
    }
    ws[OFF_VAL + i] = -softplusf(acc);
    // util first-layer halves
    for (int h = 0; h < HU; ++h) {
        float r = 0.f, cc = 0.f;
#pragma unroll
        for (int k = 0; k < EMBD; ++k) {
            r  += yg[k] * uW1[k * HU + h];
            cc += yg[k] * uW1[(EMBD + k) * HU + h];
        }
        ws[OFF_UR + i * HU + h] = r;
        ws[OFF_UC + i * HU + h] = cc;
    }
    // ext first-layer halves (pi part handled in k_pair)
    for (int h = 0; h < HE; ++h) {
        float r = 0.f, cc = 0.f;
#pragma unroll
        for (int k = 0; k < EMBD; ++k) {
            r  += yl[k] * eW1[k * HE + h];
            cc += yl[k] * eW1[(EMBD + k) * HE + h];
        }
        ws[OFF_ER + i * HE + h] = r;
        ws[OFF_EC + i * HE + h] = cc;
    }
}

// ---------------- Kernel 5: adjacency-gated pairwise util/ext + assemble ----------------
__global__ void k_pair(const float* __restrict__ adj, const float* __restrict__ pi,
                       const float* __restrict__ ub1, const float* __restrict__ uW2,
                       const float* __restrict__ ub2,
                       const float* __restrict__ eW1, const float* __restrict__ eb1,
                       const float* __restrict__ eW2, const float* __restrict__ eb2,
                       float* __restrict__ out, const float* __restrict__ ws) {
    const int i   = blockIdx.x;
    const int tid = threadIdx.x;
    __shared__ float s_ur[HU], s_ub1[HU], s_uW2[HU];
    __shared__ float s_er[HE], s_eb1[HE], s_eW2[HE], s_w0[HE], s_w1[HE];
    __shared__ float s_val[NN];

    for (int h = tid; h < HU; h += 256) {
        s_ur[h]  = ws[OFF_UR + i * HU + h];
        s_ub1[h] = ub1[h];
        s_uW2[h] = uW2[h];
    }
    for (int h = tid; h < HE; h += 256) {
        s_er[h]  = ws[OFF_ER + i * HE + h];
        s_eb1[h] = eb1[h];
        s_eW2[h] = eW2[h];
        s_w0[h]  = eW1[(2 * EMBD)     * HE + h];
        s_w1[h]  = eW1[(2 * EMBD + 1) * HE + h];
    }
    for (int j = tid; j < NN; j += 256) s_val[j] = ws[OFF_VAL + j];
    __syncthreads();

    const float vi   = s_val[i];
    const float ub2v = ub2[0];
    const float eb2v = eb2[0];

    for (int j = tid; j < NN; j += 256) {
        const float a    = adj[(size_t)i * NN + j];
        const float base = GAMMA * s_val[j] - vi;
        float o0 = base, o1 = base;
        if (a > 0.f) {
            const float* ucj = ws + OFF_UC + (size_t)j * HU;
            float au = ub2v;
#pragma unroll 4
            for (int h = 0; h < HU; ++h)
                au += reluf(s_ur[h] + ucj[h] + s_ub1[h]) * s_uW2[h];
            const float util = -softplusf(au) * a;

            const float* ecj = ws + OFF_EC + (size_t)j * HE;
            float res[2];
#pragma unroll
            for (int b = 0; b < 2; ++b) {
                const size_t pidx = (((size_t)b * NN + i) * NN + j) * 2;
                const float p0 = pi[pidx], p1 = pi[pidx + 1];
                float ae = eb2v;
#pragma unroll 4
                for (int h = 0; h < HE; ++h)
                    ae += reluf(s_er[h] + ecj[h] + p0 * s_w0[h] + p1 * s_w1[h] + s_eb1[h]) * s_eW2[h];
                res[b] = util + (-softplusf(ae)) * a + base;   // EXT_COEFF = 1.0
            }
            o0 = res[0]; o1 = res[1];
        }
        out[(size_t)i * NN + j]                    = o0;
        out[(size_t)NN * NN + (size_t)i * NN + j]  = o1;
    }
}

extern "C" void kernel_launch(void* const* d_in, const int* in_sizes, int n_in,
                              void* d_out, int out_size, void* d_ws, size_t ws_size,
                              hipStream_t stream) {
    const float* x   = (const float*)d_in[0];
    // d_in[1] = bs (int, ==2), d_in[3] = i (int) — compile-time constants here
    const float* pi  = (const float*)d_in[2];
    const float* adj = (const float*)d_in[4];
    const float* Wg  = (const float*)d_in[5];
    const float* asg = (const float*)d_in[6];
    const float* adg = (const float*)d_in[7];
    const float* Wl  = (const float*)d_in[8];
    const float* asl = (const float*)d_in[9];
    const float* adl = (const float*)d_in[10];
    const float* uW1 = (const float*)d_in[11];
    const float* ub1 = (const float*)d_in[12];
    const float* uW2 = (const float*)d_in[13];
    const float* ub2 = (const float*)d_in[14];
    const float* eW1 = (const float*)d_in[15];
    const float* eb1 = (const float*)d_in[16];
    const float* eW2 = (const float*)d_in[17];
    const float* eb2 = (const float*)d_in[18];
    const float* vW1 = (const float*)d_in[19];
    const float* vb1 = (const float*)d_in[20];
    const float* vW2 = (const float*)d_in[21];
    const float* vb2 = (const float*)d_in[22];
    float* out = (float*)d_out;
    float* ws  = (float*)d_ws;

    k_proj<<<dim3((NN + 255) / 256), 256, 0, stream>>>(x, Wg, asg, adg, Wl, asl, adl, ws);
    k_att<<<dim3(NN, 2), 256, 0, stream>>>(adj, ws);
    k_gemm_wmma<<<dim3(NN / 16, EMBD / 16, 2), 32, 0, stream>>>(ws);
    k_heads<<<dim3((NN + 255) / 256), 256, 0, stream>>>(uW1, eW1, vW1, vb1, vW2, vb2, ws);
    k_pair<<<dim3(NN), 256, 0, stream>>>(adj, pi, ub1, uW2, ub2, eW1, eb1, eW2, eb2, out, ws);
}